// MyModel_41781441856078
// MI455X (gfx1250) — compile-verified
//
#include <hip/hip_runtime.h>
#include <math.h>

// ---- problem constants (from reference) ------------------------------------
#define C_CH      144          // channels
#define W_IN      77           // batch width (GEMV K for fc)
#define KW        10           // conv kernel width
#define FC        64           // fc outputs per channel
#define CONV_OUT  68           // W_IN - KW + 1
#define CONCAT_N  (C_CH*FC + CONV_OUT)   // 9284  (divisible by 4)
#define H2        128
#define K2_STEPS  (CONCAT_N/4)           // 2321 wmma k-steps for w2 GEMV
#define KSPLIT    32                     // split-K blocks for w2
#define STEPS_PER_BLK ((K2_STEPS + KSPLIT - 1)/KSPLIT)  // 73

typedef __attribute__((ext_vector_type(2))) float v2f;
typedef __attribute__((ext_vector_type(8))) float v8f;

// One fp32 WMMA accumulate step: D(16x16) = A(16x4) x B(4x16) + C.
// 8 args: (neg_a, A, neg_b, B, c_mod, C, reuse_a, reuse_b)
__device__ __forceinline__ v8f wmma_step(v8f acc, v2f a, v2f b) {
    return __builtin_amdgcn_wmma_f32_16x16x4_f32(
        /*neg_a=*/false, a, /*neg_b=*/false, b,
        /*c_mod=*/(short)0, acc, /*reuse_a=*/false, /*reuse_b=*/false);
}

// ---------------------------------------------------------------------------
// Kernel 1: blocks [0,144) do the per-channel fc GEMVs via WMMA;
//           blocks [144,212) compute the 68 conv outputs.
// WMMA GEMV mapping (wave32, EXEC all-ones):
//   A tile:  lanes 0-15 hold rows M=lane, K=kb+0/kb+1 in a.x/a.y;
//            lanes 16-31 hold rows M=lane-16, K=kb+2/kb+3.
//   B tile:  x broadcast across all 16 N columns -> every D column = GEMV.
//   D:       lane 0 holds y[r0+0..7] in acc[0..7], lane 16 holds y[r0+8..15].
// ---------------------------------------------------------------------------
__global__ __launch_bounds__(128) void k_fc_conv(
    const float* __restrict__ x, const float* __restrict__ fc_w,
    const float* __restrict__ fc_b, const float* __restrict__ conv_w,
    const float* __restrict__ conv_b, float* __restrict__ concat)
{
    if (blockIdx.x < C_CH) {
        const int c    = blockIdx.x;
        const int lane = threadIdx.x & 31;
        const int wave = threadIdx.x >> 5;     // 0..3 -> row tile 16*wave
        const int half = lane >> 4;            // 0: K pair {0,1}, 1: {2,3}
        const int m    = lane & 15;
        const int row  = 16*wave + m;          // 0..63
        const int koff = half * 2;

        const float* wr = fc_w + ((size_t)c * FC + row) * W_IN;
        const float* xc = x + (size_t)c * W_IN;

        v8f acc = {};
        #pragma unroll 4
        for (int kb = 0; kb + 4 <= W_IN; kb += 4) {     // 19 full steps
            v2f a, b;
            a.x = wr[kb + koff];     a.y = wr[kb + koff + 1];
            b.x = xc[kb + koff];     b.y = xc[kb + koff + 1];
            acc = wmma_step(acc, a, b);
        }
        {   // K tail: only k==76 is valid (77 = 4*19 + 1).
            // Unconditional loads of a valid address + selects: no EXEC change.
            const int kb = (W_IN / 4) * 4;               // 76
            const float aw = wr[kb];
            const float bw = xc[kb];
            v2f a = {}, b = {};
            a.x = (koff == 0) ? aw : 0.0f;
            b.x = (koff == 0) ? bw : 0.0f;
            acc = wmma_step(acc, a, b);
        }
        if (m == 0) {   // lanes 0 and 16: column N=0 of D
            const int rbase = 16*wave + (half ? 8 : 0);
            #pragma unroll
            for (int r = 0; r < 8; ++r) {
                const int o = c * FC + rbase + r;
                concat[o] = acc[r] + fc_b[o];
            }
        }
    } else {
        // conv[j] = sum_{c,k} x[c, j+k] * conv_w[c,k] + conv_b
        const int j = blockIdx.x - C_CH;       // 0..67
        __shared__ float sd[128];
        float s = 0.0f;
        for (int i = threadIdx.x; i < C_CH * KW; i += 128) {
            const int c = i / KW, k = i % KW;
            s += x[c * W_IN + j + k] * conv_w[i];
        }
        sd[threadIdx.x] = s;
        __syncthreads();
        for (int off = 64; off > 0; off >>= 1) {
            if (threadIdx.x < off) sd[threadIdx.x] += sd[threadIdx.x + off];
            __syncthreads();
        }
        if (threadIdx.x == 0) concat[C_CH * FC + j] = sd[0] + conv_b[0];
    }
}

// ---------------------------------------------------------------------------
// Kernel 2: split-K WMMA GEMV for h_pre = w2 @ concat.
// 32 blocks x 8 waves; wave w owns rows 16w..16w+15; each block covers a
// contiguous range of 4-wide K steps and writes deterministic partials.
// ---------------------------------------------------------------------------
__global__ __launch_bounds__(256) void k_w2_partial(
    const float* __restrict__ concat, const float* __restrict__ w2,
    float* __restrict__ part)
{
    const int lane = threadIdx.x & 31;
    const int wave = threadIdx.x >> 5;     // 0..7
    const int half = lane >> 4;
    const int m    = lane & 15;
    const int row  = 16*wave + m;          // 0..127
    const int koff = half * 2;

    const float* wr = w2 + (size_t)row * CONCAT_N;

    const int s0 = blockIdx.x * STEPS_PER_BLK;
    const int s1 = min(s0 + STEPS_PER_BLK, K2_STEPS);

    v8f acc = {};
    for (int s = s0; s < s1; ++s) {
        const int k = 4*s + koff;
        v2f a, b;
        a.x = wr[k];       a.y = wr[k + 1];
        b.x = concat[k];   b.y = concat[k + 1];
        acc = wmma_step(acc, a, b);
    }
    if (m == 0) {
        const int rbase = 16*wave + (half ? 8 : 0);
        float* p = part + (size_t)blockIdx.x * H2;
        #pragma unroll
        for (int r = 0; r < 8; ++r) p[rbase + r] = acc[r];
    }
}

// ---------------------------------------------------------------------------
// Kernel 3: reduce split-K partials, bias+relu, w3 dot, relu, sigmoid.
// ---------------------------------------------------------------------------
__global__ __launch_bounds__(128) void k_head(
    const float* __restrict__ part, const float* __restrict__ b2,
    const float* __restrict__ w3, const float* __restrict__ b3,
    float* __restrict__ out)
{
    __shared__ float sd[H2];
    const int j = threadIdx.x;
    float s = 0.0f;
    for (int b = 0; b < KSPLIT; ++b) s += part[b * H2 + j];  // deterministic order
    float h = s + b2[j];
    h = h > 0.0f ? h : 0.0f;
    sd[j] = h * w3[j];
    __syncthreads();
    for (int off = 64; off > 0; off >>= 1) {
        if (j < off) sd[j] += sd[j + off];
        __syncthreads();
    }
    if (j == 0) {
        float o = sd[0] + b3[0];
        o = o > 0.0f ? o : 0.0f;
        out[0] = 1.0f / (1.0f + expf(-o));
    }
}

// ---------------------------------------------------------------------------
extern "C" void kernel_launch(void* const* d_in, const int* in_sizes, int n_in,
                              void* d_out, int out_size, void* d_ws, size_t ws_size,
                              hipStream_t stream)
{
    (void)in_sizes; (void)n_in; (void)out_size; (void)ws_size;
    const float* x      = (const float*)d_in[0];
    const float* fc_w   = (const float*)d_in[1];
    const float* fc_b   = (const float*)d_in[2];
    const float* conv_w = (const float*)d_in[3];
    const float* conv_b = (const float*)d_in[4];
    const float* w2     = (const float*)d_in[5];
    const float* b2     = (const float*)d_in[6];
    const float* w3     = (const float*)d_in[7];
    const float* b3     = (const float*)d_in[8];

    float* ws     = (float*)d_ws;
    float* concat = ws;                 // CONCAT_N floats (fully written each call)
    float* part   = ws + CONCAT_N;      // KSPLIT*H2 floats (fully written each call)

    k_fc_conv  <<<C_CH + CONV_OUT, 128, 0, stream>>>(x, fc_w, fc_b, conv_w, conv_b, concat);
    k_w2_partial<<<KSPLIT,         256, 0, stream>>>(concat, w2, part);
    k_head     <<<1,               H2,  0, stream>>>(part, b2, w3, b3, (float*)d_out);
}